// MultiverseEncoder_4303557230664
// MI455X (gfx1250) — compile-verified
//
#include <hip/hip_runtime.h>
#include <hip/hip_bf16.h>
#include <math.h>

#define N_NODES 50000
#define E_EDGES 800000
#define ET_TOT  (E_EDGES + N_NODES)
#define BDIM 128
#define SDIM 16
#define KIN  (BDIM + SDIM)   // 144
#define HID  128
#define HEADS 4
#define HD   (HID / HEADS)   // 32
#define ETYPES 2
#define LAYERS 3
#define GDIM 256
#define BATCHES 8
#define NEG_SLOPE 0.2f
#define LDS_PAD 4            // DWORD row pad -> conflict-free ds reads

typedef float v2f __attribute__((ext_vector_type(2)));
typedef float v4f __attribute__((ext_vector_type(4)));
typedef float v8f __attribute__((ext_vector_type(8)));
typedef unsigned int v4u __attribute__((ext_vector_type(4)));
typedef unsigned int v8u __attribute__((ext_vector_type(8)));

static inline int ceil_div(int a, int b) { return (a + b - 1) / b; }

// ---------------------------------------------------------------- utilities
__global__ void k_fill(float* __restrict__ p, float v, int n) {
  int i = blockIdx.x * blockDim.x + threadIdx.x;
  if (i < n) p[i] = v;
}

__device__ __forceinline__ void atomicMaxFloat(float* addr, float v) {
  // ordered-float max via signed max / unsigned min (init must be -inf bits)
  int iv = __float_as_int(v);
  if (iv >= 0) atomicMax((int*)addr, iv);
  else         atomicMin((unsigned int*)addr, (unsigned int)iv);
}

// ---------------------------------------------------------------- concat(board, scalars)
__global__ void k_concat(const float* __restrict__ b, const float* __restrict__ s,
                         float* __restrict__ xin) {
  int i = blockIdx.x * blockDim.x + threadIdx.x;
  if (i >= N_NODES * KIN) return;
  int n = i / KIN, c = i - n * KIN;
  xin[i] = (c < BDIM) ? b[(size_t)n * BDIM + c] : s[(size_t)n * SDIM + (c - BDIM)];
}

// ---------------------------------------------------------------- WMMA fp32 GEMM + TDM
// C[M x Ncol] = A[M x K] * B[K x Ncol] + bias.
// Block = 8 waves; TDM DMAs the 16 x K A-tile into LDS once (iterate mode, one
// K-row per iteration, lds stride K+LDS_PAD to kill bank conflicts); each wave
// then computes one 16x16 tile with V_WMMA_F32_16X16X4_F32.
//   A 16x4 : lanes 0-15 M=lane, vgpr0=K0 vgpr1=K1 ; lanes 16-31 K2,K3
//   B 4x16 : lanes 0-15 N=lane, vgpr0=K0 vgpr1=K1 ; lanes 16-31 K2,K3
//   C/D    : vgpr r, lanes 0-15 -> (M=r, N=lane), lanes 16-31 -> (M=r+8, N=lane-16)
__global__ void __launch_bounds__(256)
k_gemm(const float* __restrict__ A, const float* __restrict__ B,
       const float* __restrict__ bias, float* __restrict__ C,
       int M, int K, int Ncol) {
  __shared__ float sa[16 * (KIN + LDS_PAD)];
  const int lane = threadIdx.x;       // 0..31
  const int half = lane >> 4;         // 0/1
  const int sub  = lane & 15;
  const int tileM = blockIdx.x * 16;
  const int tileN = (blockIdx.y * blockDim.y + threadIdx.y) * 16;
  const int sstr = K + LDS_PAD;       // LDS row stride (floats)

  if (threadIdx.y == 0) {
    // ---- Tensor DMA descriptor (D#): 2D tile, iterate 16 rows ----
    unsigned long long ga = (unsigned long long)(const void*)(A + (size_t)tileM * K);
    unsigned int lds = (unsigned int)(unsigned long long)(const void*)&sa[0];
    unsigned int kd  = (unsigned int)K;
    v4u g0; v8u g1; v4u g2; v4u g3;
    g0[0] = 1u;                                   // count=1, user descriptor
    g0[1] = lds;                                  // lds_addr (byte offset)
    g0[2] = (unsigned int)ga;                     // global_addr[31:0]
    g0[3] = (unsigned int)(ga >> 32) | (2u << 30); // global_addr[56:32] | type=2
    g1[0] = (2u << 16) | (1u << 19);              // data_size=4B, iterate_enable
    g1[1] = (kd & 0xFFFFu) << 16;                 // tensor_dim0[15:0] (bits 63:48)
    g1[2] = (kd >> 16) | (0x7FFFu << 16);         // tensor_dim0[31:16] | tensor_dim1 lo (huge)
    g1[3] = 0u | (kd << 16);                      // tensor_dim1 hi | tile_dim0=K
    g1[4] = 1u;                                   // tile_dim1=1 (one row/iter), tile_dim2=0
    g1[5] = kd;                                   // tensor_dim0_stride[31:0] = K
    g1[6] = 0u;                                   // stride0 hi | stride1 lo
    g1[7] = 0u;                                   // stride1 hi
    g2[0] = 0u;                                   // tensor_dim2 (unused)
    g2[1] = (unsigned int)sstr;                   // lds_addr_increment (elements)
    g2[2] = kd;                                   // global_addr_increment[31:0] (elements)
    g2[3] = (15u << 16);                          // inc hi | iterate_count=15 (16 rows)
    g3[0] = 0u; g3[1] = 0u; g3[2] = 0u; g3[3] = 0u;
    asm volatile("tensor_load_to_lds %0, %1, %2, %3"
                 :: "s"(g0), "s"(g1), "s"(g2), "s"(g3) : "memory");
    __builtin_amdgcn_s_wait_tensorcnt(0);
  }
  __syncthreads();

  const float* Bc  = B + tileN + sub;
  const float* Ars = &sa[sub * sstr];

  v8f acc = {};
  for (int k = 0; k < K; k += 4) {
    int ka = k + 2 * half;
    v2f a, b;
    a.x = Ars[ka];
    a.y = Ars[ka + 1];
    b.x = Bc[(size_t)ka * Ncol];
    b.y = Bc[(size_t)(ka + 1) * Ncol];
    acc = __builtin_amdgcn_wmma_f32_16x16x4_f32(false, a, false, b,
                                                (short)0, acc, false, false);
  }
  float bv = bias ? bias[tileN + sub] : 0.0f;
#pragma unroll
  for (int r = 0; r < 8; ++r) {
    int row = tileM + r + 8 * half;
    if (row < M) C[(size_t)row * Ncol + tileN + sub] = acc[r] + bv;
  }
}

// ---------------------------------------------------------------- LayerNorm + ELU (+residual)
__global__ void k_post(const float* __restrict__ t, const float* __restrict__ prebias,
                       const float* __restrict__ gamma, const float* __restrict__ beta,
                       const float* __restrict__ xres, float* __restrict__ xout,
                       int rows, int ncol) {
  int row = blockIdx.x * blockDim.y + threadIdx.y;
  if (row >= rows) return;
  int lane = threadIdx.x;
  int per = ncol >> 5;                 // 4 (HID) or 8 (GDIM)
  float v[8];
  const float* tr = t + (size_t)row * ncol;
  float sum = 0.f;
  for (int j = 0; j < per; ++j) {
    int c = lane * per + j;
    float x = tr[c] + (prebias ? prebias[c] : 0.f);
    v[j] = x; sum += x;
  }
  for (int msk = 16; msk >= 1; msk >>= 1) sum += __shfl_xor(sum, msk, 32);
  float mean = sum / (float)ncol;
  float var = 0.f;
  for (int j = 0; j < per; ++j) { float d = v[j] - mean; var += d * d; }
  for (int msk = 16; msk >= 1; msk >>= 1) var += __shfl_xor(var, msk, 32);
  var /= (float)ncol;
  float inv = rsqrtf(var + 1e-5f);
  for (int j = 0; j < per; ++j) {
    int c = lane * per + j;
    float y = (v[j] - mean) * inv * gamma[c] + beta[c];
    y = (y > 0.f) ? y : expm1f(y);
    if (xres) y += xres[(size_t)row * ncol + c];
    xout[(size_t)row * ncol + c] = y;
  }
}

// ---------------------------------------------------------------- degree + loop_attr
__global__ void k_degloop(const int* __restrict__ dst, const int* __restrict__ et,
                          float* __restrict__ deg, float* __restrict__ la) {
  int e = blockIdx.x * blockDim.x + threadIdx.x;
  if (e >= E_EDGES) return;
  int d = dst[e];
  atomicAdd(&deg[d], 1.0f);
  atomicAdd(&la[(size_t)d * ETYPES + et[e]], 1.0f);
}

__global__ void k_loopnorm(float* __restrict__ la, const float* __restrict__ deg) {
  int n = blockIdx.x * blockDim.x + threadIdx.x;
  if (n >= N_NODES) return;
  float d = fmaxf(deg[n], 1.0f);
  la[2 * n]     /= d;
  la[2 * n + 1] /= d;
}

// ---------------------------------------------------------------- attention scores + segment max
// one wave per edge (E real edges then N self-loops); 4 channels/lane, head = lane/8
__global__ void k_scores(const float* __restrict__ xl, const float* __restrict__ xr,
                         const float* __restrict__ We, const float* __restrict__ att,
                         const float* __restrict__ la,
                         const int* __restrict__ src, const int* __restrict__ dst,
                         const int* __restrict__ et,
                         float* __restrict__ s, float* __restrict__ m) {
  int e = blockIdx.x * blockDim.y + threadIdx.y;
  if (e >= ET_TOT) return;
  int lane = threadIdx.x;
  int c0 = lane * 4;
  int sn, dn;
  v4f ep;
  if (e < E_EDGES) {
    sn = src[e]; dn = dst[e];
    ep = *(const v4f*)(We + (size_t)et[e] * HID + c0);
  } else {
    int n = e - E_EDGES; sn = n; dn = n;
    float l0 = la[2 * n], l1 = la[2 * n + 1];
    v4f w0 = *(const v4f*)(We + c0);
    v4f w1 = *(const v4f*)(We + HID + c0);
    ep = l0 * w0 + l1 * w1;
  }
  int h = lane >> 3;
  v4f vl = *(const v4f*)(xl + (size_t)sn * HID + c0);   // global_load_b128 gather
  v4f vr = *(const v4f*)(xr + (size_t)dn * HID + c0);
  v4f av = *(const v4f*)(att + h * HD + (lane & 7) * 4);
  float part = 0.f;
#pragma unroll
  for (int j = 0; j < 4; ++j) {
    float v = vl[j] + vr[j] + ep[j];
    v = (v > 0.f) ? v : NEG_SLOPE * v;
    part += v * av[j];
  }
  part += __shfl_xor(part, 1, 32);
  part += __shfl_xor(part, 2, 32);
  part += __shfl_xor(part, 4, 32);
  if ((lane & 7) == 0) {
    s[(size_t)e * HEADS + h] = part;
    atomicMaxFloat(&m[(size_t)dn * HEADS + h], part);
  }
}

// ---------------------------------------------------------------- exp + segment sum
__global__ void k_expsum(const int* __restrict__ dst, float* __restrict__ s,
                         const float* __restrict__ m, float* __restrict__ den) {
  int i = blockIdx.x * blockDim.x + threadIdx.x;
  if (i >= ET_TOT * HEADS) return;
  int e = i >> 2, h = i & 3;
  int dn = (e < E_EDGES) ? dst[e] : (e - E_EDGES);
  float ex = expf(s[i] - m[(size_t)dn * HEADS + h]);
  s[i] = ex;
  atomicAdd(&den[(size_t)dn * HEADS + h], ex);
}

// ---------------------------------------------------------------- alpha-weighted scatter add
__global__ void k_agg(const float* __restrict__ xl, const float* __restrict__ s,
                      const float* __restrict__ den,
                      const int* __restrict__ src, const int* __restrict__ dst,
                      float* __restrict__ out) {
  int e = blockIdx.x * blockDim.y + threadIdx.y;
  if (e >= ET_TOT) return;
  int lane = threadIdx.x;
  int sn, dn;
  if (e < E_EDGES) { sn = src[e]; dn = dst[e]; } else { sn = dn = e - E_EDGES; }
  int h = lane >> 3;
  float alpha = s[(size_t)e * HEADS + h] / den[(size_t)dn * HEADS + h];
  int c0 = lane * 4;
  v4f vl = *(const v4f*)(xl + (size_t)sn * HID + c0);   // global_load_b128 gather
  float* o = out + (size_t)dn * HID + c0;
#pragma unroll
  for (int j = 0; j < 4; ++j) atomicAdd(&o[j], vl[j] * alpha);
}

// ---------------------------------------------------------------- global pooling
__global__ void k_cnt(const int* __restrict__ batch, float* __restrict__ cnt) {
  int n = blockIdx.x * blockDim.x + threadIdx.x;
  if (n >= N_NODES) return;
  atomicAdd(&cnt[batch[n]], 1.0f);
}

__global__ void k_pool(const float* __restrict__ x, const int* __restrict__ batch,
                       float* __restrict__ gsum) {
  int i = blockIdx.x * blockDim.x + threadIdx.x;
  if (i >= N_NODES * HID) return;
  int n = i >> 7, c = i & 127;
  atomicAdd(&gsum[(size_t)batch[n] * HID + c], x[i]);
}

__global__ void k_pooldiv(const float* __restrict__ gsum, const float* __restrict__ cnt,
                          float* __restrict__ gmean) {
  int i = blockIdx.x * blockDim.x + threadIdx.x;
  if (i >= BATCHES * HID) return;
  int b = i >> 7;
  gmean[i] = gsum[i] / fmaxf(cnt[b], 1.0f);
}

// ================================================================ launch
extern "C" void kernel_launch(void* const* d_in, const int* in_sizes, int n_in,
                              void* d_out, int out_size, void* d_ws, size_t ws_size,
                              hipStream_t stream) {
  (void)in_sizes; (void)n_in; (void)out_size; (void)ws_size;
  // flattening order: top-level dict insertion order; params pytree with keys sorted
  const float* board = (const float*)d_in[0];
  const float* scal  = (const float*)d_in[1];
  const int*   eidx  = (const int*)d_in[2];   // [2][E] : src then dst
  const int*   etype = (const int*)d_in[3];
  const int*   batch = (const int*)d_in[4];
  const float* glob_W  = (const float*)d_in[5];
  const float* glob_b  = (const float*)d_in[6];
  const float* glob_be = (const float*)d_in[7];
  const float* glob_g  = (const float*)d_in[8];
  struct Layer { const float *We,*Wl,*Wr,*att,*bias,*bl,*br,*ln_b,*ln_g; } L[LAYERS];
  for (int l = 0; l < LAYERS; ++l) {
    int o = 9 + 9 * l;    // sorted: We, Wl, Wr, att, bias, bl, br, ln_b, ln_g
    L[l].We   = (const float*)d_in[o + 0];
    L[l].Wl   = (const float*)d_in[o + 1];
    L[l].Wr   = (const float*)d_in[o + 2];
    L[l].att  = (const float*)d_in[o + 3];
    L[l].bias = (const float*)d_in[o + 4];
    L[l].bl   = (const float*)d_in[o + 5];
    L[l].br   = (const float*)d_in[o + 6];
    L[l].ln_b = (const float*)d_in[o + 7];
    L[l].ln_g = (const float*)d_in[o + 8];
  }
  const float* proj_W  = (const float*)d_in[36];
  const float* proj_b  = (const float*)d_in[37];
  const float* proj_be = (const float*)d_in[38];
  const float* proj_g  = (const float*)d_in[39];

  const int* src = eidx;
  const int* dst = eidx + E_EDGES;

  float* x    = (float*)d_out;                    // N x HID node embeds (residual stream)
  float* gout = x + (size_t)N_NODES * HID;        // B x GDIM

  float* w   = (float*)d_ws;
  float* xin = w; w += (size_t)N_NODES * KIN;     // dead after proj; aliased by s/m/den
  float* xl  = w; w += (size_t)N_NODES * HID;
  float* xr  = w; w += (size_t)N_NODES * HID;
  float* tmp = w; w += (size_t)N_NODES * HID;     // GEMM result / message accumulator
  float* la  = w; w += (size_t)N_NODES * ETYPES;
  float* deg = w; w += N_NODES;
  float* gsum  = w; w += BATCHES * HID;
  float* cnt   = w; w += BATCHES;
  float* gmean = w; w += BATCHES * HID;
  float* tg    = w; w += BATCHES * GDIM;
  // aliases into xin (reused after projection):
  float* s   = xin;                               // ET*HEADS
  float* m   = xin + (size_t)ET_TOT * HEADS;      // N*HEADS
  float* den = m + (size_t)N_NODES * HEADS;       // N*HEADS

  dim3 w8(32, 8);

  // ---- degree / loop attr
  k_fill<<<ceil_div(N_NODES * ETYPES, 256), 256, 0, stream>>>(la, 0.f, N_NODES * ETYPES);
  k_fill<<<ceil_div(N_NODES, 256), 256, 0, stream>>>(deg, 0.f, N_NODES);
  k_degloop<<<ceil_div(E_EDGES, 256), 256, 0, stream>>>(dst, etype, deg, la);
  k_loopnorm<<<ceil_div(N_NODES, 256), 256, 0, stream>>>(la, deg);

  // ---- projection: x = elu(LN(concat @ W + b))
  k_concat<<<ceil_div(N_NODES * KIN, 256), 256, 0, stream>>>(board, scal, xin);
  k_gemm<<<dim3(ceil_div(N_NODES, 16), 1), w8, 0, stream>>>(xin, proj_W, proj_b, tmp,
                                                            N_NODES, KIN, HID);
  k_post<<<ceil_div(N_NODES, 8), w8, 0, stream>>>(tmp, nullptr, proj_g, proj_be,
                                                  nullptr, x, N_NODES, HID);

  // ---- GAT layers
  for (int l = 0; l < LAYERS; ++l) {
    k_gemm<<<dim3(ceil_div(N_NODES, 16), 1), w8, 0, stream>>>(x, L[l].Wl, L[l].bl, xl,
                                                              N_NODES, HID, HID);
    k_gemm<<<dim3(ceil_div(N_NODES, 16), 1), w8, 0, stream>>>(x, L[l].Wr, L[l].br, xr,
                                                              N_NODES, HID, HID);
    k_fill<<<ceil_div(N_NODES * HEADS, 256), 256, 0, stream>>>(m, -INFINITY, N_NODES * HEADS);
    k_fill<<<ceil_div(N_NODES * HEADS, 256), 256, 0, stream>>>(den, 0.f, N_NODES * HEADS);
    k_fill<<<ceil_div(N_NODES * HID, 256), 256, 0, stream>>>(tmp, 0.f, N_NODES * HID);
    k_scores<<<ceil_div(ET_TOT, 8), w8, 0, stream>>>(xl, xr, L[l].We, L[l].att, la,
                                                     src, dst, etype, s, m);
    k_expsum<<<ceil_div(ET_TOT * HEADS, 256), 256, 0, stream>>>(dst, s, m, den);
    k_agg<<<ceil_div(ET_TOT, 8), w8, 0, stream>>>(xl, s, den, src, dst, tmp);
    k_post<<<ceil_div(N_NODES, 8), w8, 0, stream>>>(tmp, L[l].bias, L[l].ln_g, L[l].ln_b,
                                                    x, x, N_NODES, HID);
  }

  // ---- global head
  k_fill<<<1, 256, 0, stream>>>(gsum, 0.f, BATCHES * HID);
  k_fill<<<1, 32, 0, stream>>>(cnt, 0.f, BATCHES);
  k_cnt<<<ceil_div(N_NODES, 256), 256, 0, stream>>>(batch, cnt);
  k_pool<<<ceil_div(N_NODES * HID, 256), 256, 0, stream>>>(x, batch, gsum);
  k_pooldiv<<<ceil_div(BATCHES * HID, 256), 256, 0, stream>>>(gsum, cnt, gmean);
  k_gemm<<<dim3(1, 2), w8, 0, stream>>>(gmean, glob_W, glob_b, tg, BATCHES, HID, GDIM);
  k_post<<<1, w8, 0, stream>>>(tg, nullptr, glob_g, glob_be, nullptr, gout, BATCHES, GDIM);
}